// VectorQuantizerEMAProject1D_38654705664776
// MI455X (gfx1250) — compile-verified
//
#include <hip/hip_runtime.h>

// ---- WMMA helpers (fp32, 16x16x4) -----------------------------------------
typedef float v2f __attribute__((ext_vector_type(2)));
typedef float v8f __attribute__((ext_vector_type(8)));

__device__ __forceinline__ v8f wmma4(v2f a, v2f b, v8f c) {
  // (neg_a, A, neg_b, B, c_mod, C, reuse_a, reuse_b)
  return __builtin_amdgcn_wmma_f32_16x16x4_f32(false, a, false, b, (short)0, c,
                                               false, false);
}

// Problem constants
#define BB 16
#define DD 256
#define TT 4096
#define PP 64
#define KK 1024
#define NN 65536  // BB*TT

// ---- Generic GEMM: out[b][m][t] = sum_k W[m][k]*X[b][k][t] + bias[m] ------
// One wave -> 16 (m) x 128 (t) strip (8 accumulators). 256 threads = 8 waves.
// mi is the fastest-varying wave index so all m-tiles sharing an X tile are
// in consecutive waves: one workgroup for M=64, two for M=256 (X streamed from HBM exactly once).
__global__ __launch_bounds__(256) void gemm_wmma(
    const float* __restrict__ W, const float* __restrict__ X,
    const float* __restrict__ bias, float* __restrict__ out, int M, int Kin) {
  const int lane = threadIdx.x & 31;
  const int wv = blockIdx.x * 8 + (threadIdx.x >> 5);
  const int mtiles = M >> 4;
  const int tc = TT >> 7;  // 32 t-groups of 128 cols
  const int perB = mtiles * tc;
  const int b = wv / perB;
  const int rem = wv - b * perB;
  const int mi = rem % mtiles;  // fastest: co-resident waves share X tile
  const int ti = rem / mtiles;

  const int col = lane & 15;           // A row index / B col index
  const int koff = (lane >> 4) << 1;   // 0 for lanes 0-15, 2 for 16-31
  const int t0 = ti << 7;

  const float* Wr = W + (size_t)(mi * 16 + col) * Kin + koff;
  const float* Xb = X + ((size_t)b * Kin + koff) * TT + t0 + col;

  v8f a0 = {}, a1 = {}, a2 = {}, a3 = {};
  v8f a4 = {}, a5 = {}, a6 = {}, a7 = {};
#pragma unroll 4
  for (int k = 0; k < Kin; k += 4) {
    v2f a;
    a.x = Wr[k];
    a.y = Wr[k + 1];
    const float* xp = Xb + (size_t)k * TT;
    v2f b0, b1, b2, b3, b4, b5, b6, b7;
    b0.x = xp[0];   b0.y = xp[TT];
    b1.x = xp[16];  b1.y = xp[TT + 16];
    b2.x = xp[32];  b2.y = xp[TT + 32];
    b3.x = xp[48];  b3.y = xp[TT + 48];
    b4.x = xp[64];  b4.y = xp[TT + 64];
    b5.x = xp[80];  b5.y = xp[TT + 80];
    b6.x = xp[96];  b6.y = xp[TT + 96];
    b7.x = xp[112]; b7.y = xp[TT + 112];
    a0 = wmma4(a, b0, a0);
    a1 = wmma4(a, b1, a1);
    a2 = wmma4(a, b2, a2);
    a3 = wmma4(a, b3, a3);
    a4 = wmma4(a, b4, a4);
    a5 = wmma4(a, b5, a5);
    a6 = wmma4(a, b6, a6);
    a7 = wmma4(a, b7, a7);
  }
  const int half = (lane >> 4) * 8;
#pragma unroll
  for (int r = 0; r < 8; ++r) {
    const int m = mi * 16 + r + half;
    const float bz = bias[m];
    float* o = out + ((size_t)b * M + m) * TT + t0 + col;
    o[0]   = a0[r] + bz;
    o[16]  = a1[r] + bz;
    o[32]  = a2[r] + bz;
    o[48]  = a3[r] + bz;
    o[64]  = a4[r] + bz;
    o[80]  = a5[r] + bz;
    o[96]  = a6[r] + bz;
    o[112] = a7[r] + bz;
  }
}

// ---- BatchNorm (train mode): per-channel stats over (B, T) ----------------
__global__ __launch_bounds__(256) void bn_stats(const float* __restrict__ Y,
                                                float* __restrict__ mean,
                                                float* __restrict__ rstd,
                                                int CH) {
  const int c = blockIdx.x;
  float s = 0.f, ss = 0.f;
  for (int b = 0; b < BB; ++b) {
    const float* p = Y + ((size_t)b * CH + c) * TT;
    for (int t = threadIdx.x; t < TT; t += 256) {
      float v = p[t];
      s += v;
      ss += v * v;
    }
  }
  __shared__ float sm[256], sq[256];
  sm[threadIdx.x] = s;
  sq[threadIdx.x] = ss;
  __syncthreads();
  for (int o = 128; o > 0; o >>= 1) {
    if (threadIdx.x < o) {
      sm[threadIdx.x] += sm[threadIdx.x + o];
      sq[threadIdx.x] += sq[threadIdx.x + o];
    }
    __syncthreads();
  }
  if (threadIdx.x == 0) {
    const float inv = 1.f / (float)NN;
    const float m = sm[0] * inv;
    const float var = sq[0] * inv - m * m;
    mean[c] = m;
    rstd[c] = rsqrtf(var + 1e-5f);
  }
}

__global__ __launch_bounds__(256) void bn_apply(float* __restrict__ Y,
                                                const float* __restrict__ mean,
                                                const float* __restrict__ rstd,
                                                const float* __restrict__ g,
                                                const float* __restrict__ bet,
                                                int CH) {
  const size_t i = ((size_t)blockIdx.x * 256 + threadIdx.x) * 4;
  const int c = (int)(i >> 12) % CH;  // i / TT % CH (TT=4096)
  const float sc = rstd[c] * g[c];
  const float sh = bet[c] - mean[c] * sc;
  float4 v = *(const float4*)(Y + i);
  v.x = v.x * sc + sh;
  v.y = v.y * sc + sh;
  v.z = v.z * sc + sh;
  v.w = v.w * sc + sh;
  *(float4*)(Y + i) = v;
}

// ---- Codebook row norms ----------------------------------------------------
__global__ __launch_bounds__(256) void emb_norms(const float* __restrict__ emb,
                                                 float* __restrict__ enorm) {
  const int k = blockIdx.x * 256 + threadIdx.x;
  if (k < KK) {
    const float* e = emb + (size_t)k * PP;
    float s = 0.f;
#pragma unroll 8
    for (int p = 0; p < PP; ++p) s += e[p] * e[p];
    enorm[k] = s;
  }
}

// ---- VQ nearest-code search via WMMA: score = ||e||^2 - 2*flat.e ----------
// One wave handles 16 rows (n); codebook staged in LDS 256 codes at a time.
__global__ __launch_bounds__(256) void vq_argmin(const float* __restrict__ Y,
                                                 const float* __restrict__ emb,
                                                 const float* __restrict__ enorm,
                                                 int* __restrict__ idx) {
  __shared__ float sE[64 * 257];  // [p][kk], pitch 257 avoids bank conflicts
  __shared__ float sN[256];
  const int tid = threadIdx.x;
  const int lane = tid & 31;
  const int wave = tid >> 5;
  const int n0 = (blockIdx.x * 8 + wave) * 16;
  const int b = n0 >> 12;      // n0 / TT
  const int t0 = n0 & 4095;    // n0 % TT
  const int row = lane & 15;
  const int koff = (lane >> 4) << 1;

  // Preload A operand: flat[n0+row][p] = Y[b][p][t0+row] for all 64 p
  float aX[16], aY[16];
#pragma unroll
  for (int s = 0; s < 16; ++s) {
    const int p0 = 4 * s + koff;
    aX[s] = Y[((size_t)b * PP + p0) * TT + t0 + row];
    aY[s] = Y[((size_t)b * PP + p0 + 1) * TT + t0 + row];
  }

  float bv[8];
  int bi[8];
#pragma unroll
  for (int r = 0; r < 8; ++r) {
    bv[r] = 3.4e38f;
    bi[r] = 0;
  }

  for (int ch = 0; ch < 4; ++ch) {
    const int kc0 = ch * 256;
    __syncthreads();
    // Stage transposed codebook chunk: sE[p][kk] = emb[kc0+kk][p]
    for (int i = tid; i < 64 * 256; i += 256) {
      const int kk = i >> 6;
      const int p = i & 63;
      sE[p * 257 + kk] = emb[(size_t)(kc0 + kk) * PP + p];
    }
    sN[tid] = enorm[kc0 + tid];
    __syncthreads();

    for (int kt = 0; kt < 16; ++kt) {
      v8f acc = {};
#pragma unroll
      for (int s = 0; s < 16; ++s) {
        const int p0 = 4 * s + koff;
        v2f a, bb;
        a.x = aX[s];
        a.y = aY[s];
        bb.x = sE[p0 * 257 + kt * 16 + row];
        bb.y = sE[(p0 + 1) * 257 + kt * 16 + row];
        acc = wmma4(a, bb, acc);
      }
      const int kg = kc0 + kt * 16 + row;
      const float en = sN[kt * 16 + row];
#pragma unroll
      for (int r = 0; r < 8; ++r) {
        const float sc = en - 2.0f * acc[r];
        if (sc < bv[r]) {  // kg strictly increases -> first-min tie rule kept
          bv[r] = sc;
          bi[r] = kg;
        }
      }
    }
  }

  // Reduce across the 16 columns within each half-wave (rows 0-7 / 8-15)
#pragma unroll
  for (int off = 1; off < 16; off <<= 1) {
#pragma unroll
    for (int r = 0; r < 8; ++r) {
      const float ov = __shfl_xor(bv[r], off, 32);
      const int oi = __shfl_xor(bi[r], off, 32);
      if (ov < bv[r] || (ov == bv[r] && oi < bi[r])) {
        bv[r] = ov;
        bi[r] = oi;
      }
    }
  }
  if (row == 0) {
    const int rbase = n0 + ((lane >> 4) << 3);
#pragma unroll
    for (int r = 0; r < 8; ++r) idx[rbase + r] = bi[r];
  }
}

// ---- Gather quantized codes back into [B][P][T] layout --------------------
__global__ __launch_bounds__(256) void vq_gather(const float* __restrict__ emb,
                                                 const int* __restrict__ idx,
                                                 float* __restrict__ qp) {
  const int n = blockIdx.x * 256 + threadIdx.x;
  const int b = n >> 12;
  const int t = n & 4095;
  const int k = idx[n];
  const float* e = emb + (size_t)k * PP;
  float* q = qp + (size_t)b * PP * TT + t;
#pragma unroll 4
  for (int p = 0; p < PP; ++p) q[(size_t)p * TT] = e[p];
}

// ---- Deterministic codebook EMA update (gather-scan, no atomics) ----------
__global__ __launch_bounds__(256) void codebook_update(
    const float* __restrict__ Y, const int* __restrict__ idx,
    float* __restrict__ ne) {
  const int k = blockIdx.x;                // one code per block
  const int seg = threadIdx.x >> 6;        // 4 n-segments
  const int p = threadIdx.x & 63;
  __shared__ float ps[4][64];
  __shared__ int pc[4];
  float s = 0.f;
  int c = 0;
  const int nb = seg * (NN / 4), nend = nb + (NN / 4);
  for (int n = nb; n < nend; ++n) {
    if (idx[n] == k) {
      const int b = n >> 12;
      const int t = n & 4095;
      s += Y[((size_t)b * PP + p) * TT + t];
      ++c;
    }
  }
  ps[seg][p] = s;
  if (p == 0) pc[seg] = c;
  __syncthreads();
  if (threadIdx.x < 64) {
    const float dw = ps[0][p] + ps[1][p] + ps[2][p] + ps[3][p];
    const float cnt = (float)(pc[0] + pc[1] + pc[2] + pc[3]);
    const float cs =
        (cnt + 1e-5f) / ((float)NN + (float)KK * 1e-5f) * (float)NN;
    ne[(size_t)k * PP + p] = dw / cs;
  }
}

// ---- Launch ----------------------------------------------------------------
extern "C" void kernel_launch(void* const* d_in, const int* in_sizes, int n_in,
                              void* d_out, int out_size, void* d_ws,
                              size_t ws_size, hipStream_t stream) {
  const float* x = (const float*)d_in[0];
  const float* W_in = (const float*)d_in[1];
  const float* b_in = (const float*)d_in[2];
  const float* g_in = (const float*)d_in[3];
  const float* beta_in = (const float*)d_in[4];
  const float* W_out = (const float*)d_in[5];
  const float* b_out = (const float*)d_in[6];
  const float* g_out = (const float*)d_in[7];
  const float* beta_out = (const float*)d_in[8];
  const float* emb = (const float*)d_in[9];

  float* out = (float*)d_out;                    // [B*D*T] = 16,777,216
  float* y = out + (size_t)BB * DD * TT;         // [B*P*T] =  4,194,304
  float* qp = y + (size_t)BB * PP * TT;          // [B*P*T]
  float* ne = qp + (size_t)BB * PP * TT;         // [K*P]

  int* idx = (int*)d_ws;                         // 65536 ints
  float* meanIn = (float*)d_ws + NN;
  float* rstdIn = meanIn + PP;
  float* meanOut = rstdIn + PP;
  float* rstdOut = meanOut + DD;
  float* enorm = rstdOut + DD;                   // 1024 floats

  // proj_in GEMM -> y (pre-BN), then BN in place
  // waves = B * (P/16) * (T/128) = 16*4*32 = 2048 -> 256 blocks of 8 waves
  gemm_wmma<<<256, 256, 0, stream>>>(W_in, x, b_in, y, PP, DD);
  bn_stats<<<PP, 256, 0, stream>>>(y, meanIn, rstdIn, PP);
  bn_apply<<<(BB * PP * TT) / 1024, 256, 0, stream>>>(y, meanIn, rstdIn, g_in,
                                                      beta_in, PP);
  // VQ: norms, argmin (WMMA), gather, codebook update
  emb_norms<<<4, 256, 0, stream>>>(emb, enorm);
  vq_argmin<<<512, 256, 0, stream>>>(y, emb, enorm, idx);
  vq_gather<<<NN / 256, 256, 0, stream>>>(emb, idx, qp);
  codebook_update<<<KK, 256, 0, stream>>>(y, idx, ne);
  // proj_out GEMM -> out (pre-BN), then BN in place
  // waves = B * (D/16) * (T/128) = 16*16*32 = 8192 -> 1024 blocks
  gemm_wmma<<<1024, 256, 0, stream>>>(W_out, qp, b_out, out, DD, PP);
  bn_stats<<<DD, 256, 0, stream>>>(out, meanOut, rstdOut, DD);
  bn_apply<<<(BB * DD * TT) / 1024, 256, 0, stream>>>(out, meanOut, rstdOut,
                                                      g_out, beta_out, DD);
}